// DeformableConv2d_5970004541637
// MI455X (gfx1250) — compile-verified
//
#include <hip/hip_runtime.h>
#include <math.h>

// ---------------- problem constants ----------------
#define Bn   8
#define CIN  256
#define COUT 256
#define Hh   48
#define Ww   48
#define K2   9
#define HW   (Hh*Ww)        // 2304
#define NTOT (Bn*HW)        // 18432
#define KDIM (CIN*K2)       // 2304 reduction length

// ---------------- kernel1 (aux conv) config ----------------
#define CC     32           // input-channel chunk
#define AUXC   27           // 18 offset + 9 mask channels
#define PAIRS  (Ww*AUXC)    // 1296 (wo, och) pairs per (b,ho)

// ---------------- kernel2 (WMMA DCN) config ----------------
#define NT     64           // spatial tile (pixels)
#define MT     128          // Cout tile
#define CK     16           // channels per K-chunk
#define KC     (CK*K2)      // 144 K-values per chunk
#define VSTR   72           // padded Vs row stride (banks: halves disjoint)
#define NKSTEP (KC/4)       // 36 WMMA K-steps per chunk

typedef float v2f __attribute__((ext_vector_type(2)));
typedef float v8f __attribute__((ext_vector_type(8)));

// ============================================================
// Kernel 0: weight transpose -> Wtp[kpair][Cout][2] in workspace.
// Makes per-k-step WMMA A-fragments a single coalesced b64 load.
// ============================================================
__global__ __launch_bounds__(256)
void dcn_wtrans_kernel(const float* __restrict__ W, float* __restrict__ Wtp)
{
    int idx = blockIdx.x * 256 + threadIdx.x;      // over KDIM*COUT = 589824
    int j     = idx & 1;
    int o     = (idx >> 1) & (COUT - 1);
    int kpair = idx >> 9;
    // write coalesced; read scattered (one-off, L2 absorbs it)
    Wtp[idx] = W[(size_t)o * KDIM + kpair * 2 + j];
}

// ============================================================
// Kernel 1: offset/mask 3x3 conv -> py/px/mask tables in ws.
// One workgroup per (b, ho) output row.
// ============================================================
__global__ __launch_bounds__(256)
void dcn_aux_kernel(const float* __restrict__ x,
                    const float* __restrict__ off_w, const float* __restrict__ off_b,
                    const float* __restrict__ mask_w, const float* __restrict__ mask_b,
                    float* __restrict__ py_ws, float* __restrict__ px_ws,
                    float* __restrict__ m_ws)
{
    __shared__ float xs[CC][3][50];      // 3 input rows, cols -1..48 (idx col = wcol+1)
    __shared__ float wsl[AUXC * (CC*9)]; // aux-weight chunk, [och][ci*9+k]

    const int b  = blockIdx.x / Hh;
    const int ho = blockIdx.x % Hh;
    const int t  = threadIdx.x;

    float acc[6];
#pragma unroll
    for (int i = 0; i < 6; ++i) acc[i] = 0.f;

    for (int c0 = 0; c0 < CIN; c0 += CC) {
        __syncthreads();
        // ---- stage x slab: 32 ch x 3 rows x 50 cols (zero-padded) ----
        for (int idx = t; idx < CC*3*50; idx += 256) {
            int ci = idx / 150, rem = idx % 150;
            int r = rem / 50, col = rem % 50;
            int hy = ho - 1 + r, wc = col - 1;
            float v = 0.f;
            if ((unsigned)hy < (unsigned)Hh && (unsigned)wc < (unsigned)Ww)
                v = x[(((size_t)b*CIN + (c0+ci))*Hh + hy)*Ww + wc];
            xs[ci][r][col] = v;
        }
        // ---- stage aux weights: [och][ci*9+k] (contiguous per row) ----
        for (int idx = t; idx < AUXC*(CC*9); idx += 256) {
            int och = idx / (CC*9), j = idx % (CC*9);
            const float* src = (och < 18)
                ? (off_w  + (size_t)och      *(CIN*9) + c0*9 + j)
                : (mask_w + (size_t)(och-18) *(CIN*9) + c0*9 + j);
            wsl[idx] = *src;
        }
        __syncthreads();
        // ---- accumulate ----
        for (int i = 0; i < 6; ++i) {
            int p = i*256 + t;
            if (p >= PAIRS) break;
            int och = p / Ww, wo = p % Ww;
            float a = acc[i];
            const float* wrow = &wsl[och * (CC*9)];
#pragma unroll 1
            for (int ci = 0; ci < CC; ++ci) {
                const float* wp = wrow + ci*9;
                a = fmaf(wp[0], xs[ci][0][wo+0], a);
                a = fmaf(wp[1], xs[ci][0][wo+1], a);
                a = fmaf(wp[2], xs[ci][0][wo+2], a);
                a = fmaf(wp[3], xs[ci][1][wo+0], a);
                a = fmaf(wp[4], xs[ci][1][wo+1], a);
                a = fmaf(wp[5], xs[ci][1][wo+2], a);
                a = fmaf(wp[6], xs[ci][2][wo+0], a);
                a = fmaf(wp[7], xs[ci][2][wo+1], a);
                a = fmaf(wp[8], xs[ci][2][wo+2], a);
            }
            acc[i] = a;
        }
    }

    // ---- bias + transform -> sampling tables ----
    for (int i = 0; i < 6; ++i) {
        int p = i*256 + t;
        if (p >= PAIRS) break;
        int och = p / Ww, wo = p % Ww;
        int s = ho*Ww + wo;
        if (och < 18) {                       // offset channels: (dy,dx) per tap
            int k = och >> 1, isx = och & 1;
            float v = acc[i] + off_b[och];
            size_t idx = ((size_t)b*K2 + k)*HW + s;
            if (isx) px_ws[idx] = v + (float)(k % 3) + (float)wo - 1.f;
            else     py_ws[idx] = v + (float)(k / 3) + (float)ho - 1.f;
        } else {                              // mask channels
            int k = och - 18;
            float v = acc[i] + mask_b[k];
            m_ws[((size_t)b*K2 + k)*HW + s] = 1.f / (1.f + __expf(-v));
        }
    }
}

// ============================================================
// Kernel 2: deformable GEMM with V_WMMA_F32_16X16X4_F32.
// grid.x = spatial tile (NTOT/NT = 288), grid.y = Cout tile (2).
// 256 threads = 8 waves; each wave owns a 16-row Cout stripe x 64 cols.
// A fragments: one coalesced global_load_b64 from Wtp per k-step.
// B fragments: ds_load (2addr-fused) from bank-padded Vs tile.
// ============================================================
__global__ __launch_bounds__(256)
void dcn_wmma_kernel(const float* __restrict__ x, const float* __restrict__ Wtp,
                     const float* __restrict__ py_ws, const float* __restrict__ px_ws,
                     const float* __restrict__ m_ws, float* __restrict__ out)
{
    __shared__ float Vs[KC * VSTR];      // gathered V tile: [k_local][n] (padded)
    __shared__ int   offs[K2*NT*4];      // clamped corner flat offsets per (k,n)
    __shared__ float wts [K2*NT*4];      // validity * bilinear * mask per corner

    const int tile  = blockIdx.x;
    const int b     = tile / (HW/NT);
    const int s0    = (tile % (HW/NT)) * NT;
    const int mbase = blockIdx.y * MT;
    const int t     = threadIdx.x;

    // ---- per-(tap, pixel) sampling parameters (once per WG) ----
    for (int p = t; p < K2*NT; p += 256) {
        int k = p / NT, n = p % NT;
        size_t gi = ((size_t)b*K2 + k)*HW + s0 + n;
        float py = py_ws[gi], px = px_ws[gi], m = m_ws[gi];
        float y0f = floorf(py), x0f = floorf(px);
        float wy = py - y0f, wx = px - x0f;
        int y0 = (int)y0f, xq = (int)x0f;
        float bw[4] = {(1.f-wy)*(1.f-wx), (1.f-wy)*wx, wy*(1.f-wx), wy*wx};
#pragma unroll
        for (int c = 0; c < 4; ++c) {
            int yy = y0 + (c >> 1), xx = xq + (c & 1);
            bool valid = (yy >= 0) && (yy < Hh) && (xx >= 0) && (xx < Ww);
            int yc = min(max(yy, 0), Hh-1), xc = min(max(xx, 0), Ww-1);
            offs[p*4 + c] = yc*Ww + xc;
            wts [p*4 + c] = valid ? bw[c]*m : 0.f;
        }
    }
    __syncthreads();

    const int wave = t >> 5, lane = t & 31;
    const int half = lane >> 4, ml = lane & 15;
    const int m0 = mbase + wave*16;

    v8f acc[4];
#pragma unroll
    for (int j = 0; j < 4; ++j)
#pragma unroll
        for (int r = 0; r < 8; ++r) acc[j][r] = 0.f;

    // Per-lane A base in transposed weight: kpair = half, row m0+ml.
    // Per k-step advance = 2 kpairs = 1024 floats.
    const float* abase0 = Wtp + ((size_t)half * COUT + (m0 + ml)) * 2;

    for (int c0 = 0; c0 < CIN; c0 += CK) {
        // ---- fill V tile: bilinear gather (L2-hot x) ----
        const float* xc = x + ((size_t)b*CIN + c0)*HW;
        for (int p = t; p < K2*NT; p += 256) {
            int k = p / NT, n = p % NT;
            int   o0 = offs[p*4+0], o1 = offs[p*4+1], o2 = offs[p*4+2], o3 = offs[p*4+3];
            float w0 = wts [p*4+0], w1 = wts [p*4+1], w2 = wts [p*4+2], w3 = wts [p*4+3];
#pragma unroll 4
            for (int ci = 0; ci < CK; ++ci) {
                const float* bp = xc + ci*HW;
                float v = fmaf(w0, bp[o0], fmaf(w1, bp[o1], fmaf(w2, bp[o2], w3*bp[o3])));
                Vs[(ci*K2 + k)*VSTR + n] = v;
            }
        }
        __syncthreads();

        // A pointer for this chunk: global kpair base = c0*9/2
        const float* ap = abase0 + (size_t)(c0 * (K2/1) / 2) * COUT * 2;
        // prefetch next chunk's A stream
        if (c0 + CK < CIN)
            __builtin_prefetch(ap + (size_t)(KC/2) * COUT * 2, 0, 3);

        // ---- WMMA over this K-chunk ----
#pragma unroll 4
        for (int ks = 0; ks < NKSTEP; ++ks) {
            // A: K = (c0*9 + ks*4 + half*2) + {0,1}, row m0+ml  -> one b64
            v2f a = *(const v2f*)(ap + (size_t)ks * (2 * COUT * 2));
            int krow = ks*4 + half*2;
#pragma unroll
            for (int j = 0; j < 4; ++j) {
                v2f bf;
                bf.x = Vs[(krow    )*VSTR + j*16 + ml];
                bf.y = Vs[(krow + 1)*VSTR + j*16 + ml];
                acc[j] = __builtin_amdgcn_wmma_f32_16x16x4_f32(
                    /*neg_a=*/false, a, /*neg_b=*/false, bf,
                    /*c_mod=*/(short)0, acc[j], /*reuse_a=*/false, /*reuse_b=*/false);
            }
        }
        __syncthreads();
    }

    // ---- epilogue: C layout -> NCHW ----
#pragma unroll
    for (int j = 0; j < 4; ++j) {
#pragma unroll
        for (int r = 0; r < 8; ++r) {
            int row = m0 + half*8 + r;
            int col = s0 + j*16 + ml;
            out[((size_t)b*COUT + row)*HW + col] = acc[j][r];
        }
    }
}

// ============================================================
extern "C" void kernel_launch(void* const* d_in, const int* in_sizes, int n_in,
                              void* d_out, int out_size, void* d_ws, size_t ws_size,
                              hipStream_t stream) {
    const float* x      = (const float*)d_in[0];
    const float* weight = (const float*)d_in[1];
    const float* off_w  = (const float*)d_in[2];
    const float* off_b  = (const float*)d_in[3];
    const float* mask_w = (const float*)d_in[4];
    const float* mask_b = (const float*)d_in[5];
    float* out = (float*)d_out;

    // workspace layout (floats):
    //   py / px / mask tables, each [B, K2, H*W]            (~664 KB each)
    //   Wtp: transposed weight [KDIM/2][COUT][2]            (~2.36 MB)
    float* py_ws = (float*)d_ws;
    float* px_ws = py_ws + (size_t)Bn*K2*HW;
    float* m_ws  = px_ws + (size_t)Bn*K2*HW;
    float* Wtp   = m_ws  + (size_t)Bn*K2*HW;

    dcn_wtrans_kernel<<<dim3((KDIM*COUT)/256), 256, 0, stream>>>(weight, Wtp);

    dcn_aux_kernel<<<dim3(Bn*Hh), 256, 0, stream>>>(
        x, off_w, off_b, mask_w, mask_b, py_ws, px_ws, m_ws);

    dcn_wmma_kernel<<<dim3(NTOT/NT, COUT/MT), 256, 0, stream>>>(
        x, Wtp, py_ws, px_ws, m_ws, out);
}